// compute_message_9216999817568
// MI455X (gfx1250) — compile-verified
//
#include <hip/hip_runtime.h>

// SchNet-style continuous-filter conv layer, fused for MI455X (gfx1250).
//  K1: zero agg
//  K2: hv = feat @ W_in2f                       (WMMA f32 16x16x4)
//  K3: fused edge kernel: he = (swish(fij@Wf1+b)@Wf2+b)*C(r);
//      m = hv[src]*he; atomicAdd(agg[dst], m)   (WMMA + LDS relayout + f32 atomics)
//  K4: out = swish(agg@Wm1+b)@Wm2+b             (two fused WMMA GEMMs)
//
// Round-2 -> Round-3: stage all 8 B-fragments per K-step in a register array
// (batch the ds_loads into one DScnt group -> single wait before the 8-WMMA
// burst instead of load/wait/wmma serialization); hoist per-wave-invariant
// bias values out of the persistent tile loop.

typedef float v2f __attribute__((ext_vector_type(2)));
typedef float v8f __attribute__((ext_vector_type(8)));

#define N_NODES 50000
#define N_EDGES 1600000
#define NF      128      // filter dim == n_in
#define NG      50       // gaussian basis
#define NG_PAD  56       // padded K for GEMM1 (>=52, keeps b64 alignment)
#define KSTEPS1 13       // ceil(50/4)
#define WPB     4        // waves per block (128 threads)
#define CUT_R   5.0f
#define H1S     132      // padded LDS row stride (floats) to dodge bank conflicts

__device__ __forceinline__ v8f wmma_f32(v2f a, v2f b, v8f c) {
  // D = A(16x4,f32) * B(4x16,f32) + C(16x16,f32)
  return __builtin_amdgcn_wmma_f32_16x16x4_f32(false, a, false, b, (short)0, c,
                                               false, false);
}

__device__ __forceinline__ float swish_f(float x) {
  return x / (1.0f + __expf(-x));
}

// -------------------------------------------------------------------------
__global__ void zero_kernel(float* __restrict__ p, long n) {
  long i = (long)blockIdx.x * blockDim.x + threadIdx.x;
  long stride = (long)gridDim.x * blockDim.x;
  for (; i < n; i += stride) p[i] = 0.0f;
}

// -------------------------------------------------------------------------
// hv = feat @ W_in2f   ([50000,128] @ [128,128])
__global__ void __launch_bounds__(32 * WPB, 1)
in2f_kernel(const float* __restrict__ feat,
            const float* __restrict__ W,   // [128][128] row-major
            float* __restrict__ hv, int ntiles) {
  extern __shared__ float smem[];
  float* Wt = smem;  // transposed: Wt[col*H1S + k] = W[k][col], 128*132 floats
  const int tid = threadIdx.x;
  for (int idx = tid; idx < NF * H1S; idx += blockDim.x) {
    int col = idx / H1S, k = idx % H1S;
    Wt[idx] = (k < NF) ? W[k * NF + col] : 0.0f;
  }
  __syncthreads();

  const int lane = tid & 31, wave = tid >> 5;
  const int hgrp = lane >> 4;       // lane-half: K offset {0,2}
  const int mrow = lane & 15;       // A's M / B's N / D's N
  const int gw0 = blockIdx.x * WPB + wave;
  const int nwaves = gridDim.x * WPB;
  const float* bbase = Wt + mrow * H1S + 2 * hgrp;

  for (int tile = gw0; tile < ntiles; tile += nwaves) {
    const int r0 = tile * 16;
    const float* arow = feat + (long)(r0 + mrow) * NF + 2 * hgrp;
    if (tile + nwaves < ntiles)
      __builtin_prefetch(feat + (long)((tile + nwaves) * 16 + mrow) * NF, 0, 3);

    v8f acc[8];
#pragma unroll
    for (int n = 0; n < 8; ++n) acc[n] = (v8f){};

#pragma unroll 4
    for (int k = 0; k < 32; ++k) {
      v2f a = *(const v2f*)(arow + 4 * k);
      v2f b[8];
#pragma unroll
      for (int n = 0; n < 8; ++n)
        b[n] = *(const v2f*)(bbase + n * 16 * H1S + 4 * k);
#pragma unroll
      for (int n = 0; n < 8; ++n) acc[n] = wmma_f32(a, b[n], acc[n]);
    }
#pragma unroll
    for (int j = 0; j < 8; ++j) {
      int M = j + 8 * hgrp;
      float* orow = hv + (long)(r0 + M) * NF + mrow;
#pragma unroll
      for (int n = 0; n < 8; ++n) orow[n * 16] = acc[n][j];
    }
  }
}

// -------------------------------------------------------------------------
// Fused edge kernel.
__global__ void __launch_bounds__(32 * WPB, 1)
edge_kernel(const float* __restrict__ fij,
            const float* __restrict__ rij,
            const int* __restrict__ src,
            const int* __restrict__ dst,
            const float* __restrict__ Wf1,
            const float* __restrict__ bf1,
            const float* __restrict__ Wf2,
            const float* __restrict__ bf2,
            const float* __restrict__ hv,
            float* __restrict__ agg, int ntiles) {
  extern __shared__ float smem[];
  float* Wt1 = smem;                       // [128][56]  transposed+padded Wf1
  float* Wt2 = Wt1 + NF * NG_PAD;          // [128][132] transposed+padded Wf2
  float* b1s = Wt2 + NF * H1S;             // 128
  float* b2s = b1s + NF;                   // 128
  float* h1s = b2s + NF;                   // WPB * 16*132
  float* cutS = h1s + WPB * 16 * H1S;      // WPB * 16
  int* srcS = (int*)(cutS + WPB * 16);     // WPB * 16
  int* dstS = srcS + WPB * 16;             // WPB * 16

  const int tid = threadIdx.x;
  for (int idx = tid; idx < NF * NG_PAD; idx += blockDim.x) {
    int col = idx / NG_PAD, k = idx % NG_PAD;
    Wt1[idx] = (k < NG) ? Wf1[k * NF + col] : 0.0f;
  }
  for (int idx = tid; idx < NF * H1S; idx += blockDim.x) {
    int col = idx / H1S, k = idx % H1S;
    Wt2[idx] = (k < NF) ? Wf2[k * NF + col] : 0.0f;
  }
  for (int idx = tid; idx < NF; idx += blockDim.x) {
    b1s[idx] = bf1[idx];
    b2s[idx] = bf2[idx];
  }
  __syncthreads();

  const int lane = tid & 31, wave = tid >> 5;
  const int hgrp = lane >> 4, mrow = lane & 15;
  float* h1 = h1s + wave * 16 * H1S;
  float* cw = cutS + wave * 16;
  int* sw = srcS + wave * 16;
  int* dw = dstS + wave * 16;

  const float* b1base = Wt1 + mrow * NG_PAD + 2 * hgrp;
  const float* b2base = Wt2 + mrow * H1S + 2 * hgrp;
  const float* a2base = h1 + mrow * H1S + 2 * hgrp;

  // per-wave-invariant bias values (one per column tile)
  float bias1[8], bias2[8];
#pragma unroll
  for (int n = 0; n < 8; ++n) {
    bias1[n] = b1s[n * 16 + mrow];
    bias2[n] = b2s[n * 16 + mrow];
  }

  const int gw0 = blockIdx.x * WPB + wave;
  const int nwaves = gridDim.x * WPB;
  const float PI_OVER_RC = 3.14159265358979323846f / CUT_R;

  for (int tile = gw0; tile < ntiles; tile += nwaves) {
    const int e0 = tile * 16;

    if (lane < 16) {                       // per-edge scalars for this tile
      float r = rij[e0 + lane];
      cw[lane] = (r < CUT_R) ? 0.5f * (__cosf(PI_OVER_RC * r) + 1.0f) : 0.0f;
      sw[lane] = src[e0 + lane];
      dw[lane] = dst[e0 + lane];
    }
    if (tile + nwaves < ntiles)
      __builtin_prefetch(fij + (long)((tile + nwaves) * 16 + mrow) * NG, 0, 3);

    // ---- GEMM1 (+bias, swish) -> per-wave LDS tile [16][128] ----
    {
      const float* arow = fij + (long)(e0 + mrow) * NG;
      v8f acc[8];
#pragma unroll
      for (int n = 0; n < 8; ++n) acc[n] = (v8f){};
#pragma unroll
      for (int k = 0; k < KSTEPS1; ++k) {
        int kk = 4 * k + 2 * hgrp;
        v2f a;
        a.x = (kk < NG) ? arow[kk] : 0.0f;
        a.y = (kk + 1 < NG) ? arow[kk + 1] : 0.0f;
        v2f b[8];
#pragma unroll
        for (int n = 0; n < 8; ++n)
          b[n] = *(const v2f*)(b1base + n * 16 * NG_PAD + 4 * k);
#pragma unroll
        for (int n = 0; n < 8; ++n) acc[n] = wmma_f32(a, b[n], acc[n]);
      }
#pragma unroll
      for (int j = 0; j < 8; ++j) {
        int M = j + 8 * hgrp;
        float* hrow = h1 + M * H1S + mrow;
#pragma unroll
        for (int n = 0; n < 8; ++n)
          hrow[n * 16] = swish_f(acc[n][j] + bias1[n]);
      }
    }

    // ---- GEMM2 (K=128) + cutoff + gather-mul + scatter-add ----
    {
      v8f acc[8];
#pragma unroll
      for (int n = 0; n < 8; ++n) acc[n] = (v8f){};
#pragma unroll 4
      for (int k = 0; k < 32; ++k) {
        v2f av = *(const v2f*)(a2base + 4 * k);
        v2f bv[8];
#pragma unroll
        for (int n = 0; n < 8; ++n)
          bv[n] = *(const v2f*)(b2base + n * 16 * H1S + 4 * k);
#pragma unroll
        for (int n = 0; n < 8; ++n) acc[n] = wmma_f32(av, bv[n], acc[n]);
      }
#pragma unroll
      for (int j = 0; j < 8; ++j) {
        int M = j + 8 * hgrp;
        float cut = cw[M];
        const float* hvrow = hv + (long)sw[M] * NF + mrow;
        float* aggrow = agg + (long)dw[M] * NF + mrow;
#pragma unroll
        for (int n = 0; n < 8; ++n) {
          float he = (acc[n][j] + bias2[n]) * cut;
          float m = hvrow[n * 16] * he;
          unsafeAtomicAdd(aggrow + n * 16, m);  // global_atomic_add_f32
        }
      }
    }
  }
}

// -------------------------------------------------------------------------
// out = swish(agg @ Wm1 + bm1) @ Wm2 + bm2
__global__ void __launch_bounds__(32 * WPB, 1)
msg_transform_kernel(const float* __restrict__ agg,
                     const float* __restrict__ Wm1,
                     const float* __restrict__ bm1,
                     const float* __restrict__ Wm2,
                     const float* __restrict__ bm2,
                     float* __restrict__ out, int ntiles) {
  extern __shared__ float smem[];
  float* Wt1 = smem;               // [128][132]
  float* Wt2 = Wt1 + NF * H1S;     // [128][132]
  float* b1s = Wt2 + NF * H1S;
  float* b2s = b1s + NF;
  float* h1s = b2s + NF;           // WPB * 16*132

  const int tid = threadIdx.x;
  for (int idx = tid; idx < NF * H1S; idx += blockDim.x) {
    int col = idx / H1S, k = idx % H1S;
    bool ok = (k < NF);
    Wt1[idx] = ok ? Wm1[k * NF + col] : 0.0f;
    Wt2[idx] = ok ? Wm2[k * NF + col] : 0.0f;
  }
  for (int idx = tid; idx < NF; idx += blockDim.x) {
    b1s[idx] = bm1[idx];
    b2s[idx] = bm2[idx];
  }
  __syncthreads();

  const int lane = tid & 31, wave = tid >> 5;
  const int hgrp = lane >> 4, mrow = lane & 15;
  float* h1 = h1s + wave * 16 * H1S;
  const float* b1base = Wt1 + mrow * H1S + 2 * hgrp;
  const float* b2base = Wt2 + mrow * H1S + 2 * hgrp;
  const float* a2base = h1 + mrow * H1S + 2 * hgrp;

  float bias1[8], bias2[8];
#pragma unroll
  for (int n = 0; n < 8; ++n) {
    bias1[n] = b1s[n * 16 + mrow];
    bias2[n] = b2s[n * 16 + mrow];
  }

  const int gw0 = blockIdx.x * WPB + wave;
  const int nwaves = gridDim.x * WPB;

  for (int tile = gw0; tile < ntiles; tile += nwaves) {
    const int r0 = tile * 16;
    const float* arow = agg + (long)(r0 + mrow) * NF + 2 * hgrp;
    if (tile + nwaves < ntiles)
      __builtin_prefetch(agg + (long)((tile + nwaves) * 16 + mrow) * NF, 0, 3);

    // ---- GEMM1 (+bias, swish) -> per-wave LDS tile ----
    {
      v8f acc[8];
#pragma unroll
      for (int n = 0; n < 8; ++n) acc[n] = (v8f){};
#pragma unroll 4
      for (int k = 0; k < 32; ++k) {
        v2f a = *(const v2f*)(arow + 4 * k);
        v2f b[8];
#pragma unroll
        for (int n = 0; n < 8; ++n)
          b[n] = *(const v2f*)(b1base + n * 16 * H1S + 4 * k);
#pragma unroll
        for (int n = 0; n < 8; ++n) acc[n] = wmma_f32(a, b[n], acc[n]);
      }
#pragma unroll
      for (int j = 0; j < 8; ++j) {
        int M = j + 8 * hgrp;
        float* hrow = h1 + M * H1S + mrow;
#pragma unroll
        for (int n = 0; n < 8; ++n)
          hrow[n * 16] = swish_f(acc[n][j] + bias1[n]);
      }
    }

    // ---- GEMM2 (+bias) -> out ----
    {
      v8f acc[8];
#pragma unroll
      for (int n = 0; n < 8; ++n) acc[n] = (v8f){};
#pragma unroll 4
      for (int k = 0; k < 32; ++k) {
        v2f av = *(const v2f*)(a2base + 4 * k);
        v2f bv[8];
#pragma unroll
        for (int n = 0; n < 8; ++n)
          bv[n] = *(const v2f*)(b2base + n * 16 * H1S + 4 * k);
#pragma unroll
        for (int n = 0; n < 8; ++n) acc[n] = wmma_f32(av, bv[n], acc[n]);
      }
#pragma unroll
      for (int j = 0; j < 8; ++j) {
        int M = j + 8 * hgrp;
        float* orow = out + (long)(r0 + M) * NF + mrow;
#pragma unroll
        for (int n = 0; n < 8; ++n)
          orow[n * 16] = acc[n][j] + bias2[n];
      }
    }
  }
}

// -------------------------------------------------------------------------
extern "C" void kernel_launch(void* const* d_in, const int* in_sizes, int n_in,
                              void* d_out, int out_size, void* d_ws, size_t ws_size,
                              hipStream_t stream) {
  (void)in_sizes; (void)n_in; (void)out_size; (void)ws_size;
  const float* feat   = (const float*)d_in[0];
  const float* fij    = (const float*)d_in[1];
  const float* rij    = (const float*)d_in[2];
  const int*   src    = (const int*)d_in[3];
  const int*   dst    = (const int*)d_in[4];
  const float* W_in2f = (const float*)d_in[5];
  const float* Wf1    = (const float*)d_in[6];
  const float* bf1    = (const float*)d_in[7];
  const float* Wf2    = (const float*)d_in[8];
  const float* bf2    = (const float*)d_in[9];
  const float* Wm1    = (const float*)d_in[10];
  const float* bm1    = (const float*)d_in[11];
  const float* Wm2    = (const float*)d_in[12];
  const float* bm2    = (const float*)d_in[13];
  float* out = (float*)d_out;

  float* hv  = (float*)d_ws;                       // [N, 128]
  float* agg = hv + (size_t)N_NODES * NF;          // [N, 128]

  constexpr int IN2F_LDS = NF * H1S * 4;                                  // 66 KB
  constexpr int EDGE_LDS =
      (NF * NG_PAD + NF * H1S + 2 * NF + WPB * 16 * H1S + WPB * 16) * 4 +
      2 * WPB * 16 * 4;                                                   // ~129 KB
  constexpr int MSG_LDS =
      (2 * NF * H1S + 2 * NF + WPB * 16 * H1S) * 4;                       // ~166 KB

  zero_kernel<<<2048, 256, 0, stream>>>(agg, (long)N_NODES * NF);
  in2f_kernel<<<256, 32 * WPB, IN2F_LDS, stream>>>(feat, W_in2f, hv,
                                                   N_NODES / 16);
  edge_kernel<<<2048, 32 * WPB, EDGE_LDS, stream>>>(fij, rij, src, dst, Wf1,
                                                    bf1, Wf2, bf2, hv, agg,
                                                    N_EDGES / 16);
  msg_transform_kernel<<<256, 32 * WPB, MSG_LDS, stream>>>(agg, Wm1, bm1, Wm2,
                                                           bm2, out,
                                                           N_NODES / 16);
}